// SingleVQWithEMA_47012712022108
// MI455X (gfx1250) — compile-verified
//
#include <hip/hip_runtime.h>

typedef float v2f __attribute__((ext_vector_type(2)));
typedef float v8f __attribute__((ext_vector_type(8)));

#define K_CODES 4096
#define DIM     128
#define BATCH   8
#define TLEN    3000
#define NVEC    (BATCH * TLEN)      // 24000
#define NTILES  (NVEC / 16)         // 1500
#define CHUNK   64
#define LDS_STRIDE 132              // pad: 132 % 64 = 4 -> conflict-free B fetch

// ---- workspace layout (float offsets) ----
#define WS_IDX    0                 // int[NVEC]
#define WS_NRM    24000             // float[K]
#define WS_CNT    28096             // float[K]
#define WS_SUM    32192             // float[K*D]
#define WS_CSPD   556480            // float[K]  (pre-dead new_cs)
#define WS_RAND   560576            // int[K]
#define WS_NSUM   564672            // float[1]
#define WS_LOSSP  564673            // float[2*NTILES]

// ---- output layout (float offsets, tuple order) ----
#define OUT_ZQ    0
#define OUT_CODES 3072000
#define OUT_LOSS  3096000
#define OUT_CB    3096001
#define OUT_CS    3620289
#define OUT_EA    3624385

// ---------------- zero scratch ----------------
__global__ void zero_ws_kernel(float* ws) {
  int gid = blockIdx.x * 256 + threadIdx.x;
  if (gid < K_CODES + K_CODES * DIM) ws[WS_CNT + gid] = 0.0f;  // counts + sums
  if (gid == 0) ws[WS_NSUM] = 0.0f;
}

// ---------------- codebook row norms (one wave per code) ----------------
__global__ void norms_kernel(const float* __restrict__ cb, float* __restrict__ ws) {
  int wave = threadIdx.x >> 5, lane = threadIdx.x & 31;
  int code = blockIdx.x * 4 + wave;
  const float* row = cb + (size_t)code * DIM;
  float s = 0.0f;
#pragma unroll
  for (int j = 0; j < 4; ++j) { float v = row[lane + j * 32]; s += v * v; }
#pragma unroll
  for (int off = 16; off; off >>= 1) s += __shfl_xor(s, off, 32);
  if (lane == 0) ws[WS_NRM + code] = s;
}

// ---------------- main VQ kernel: fp32 WMMA distance GEMM + argmin ----------------
__global__ void __launch_bounds__(128)
vq_kernel(const float* __restrict__ z, const float* __restrict__ cb,
          float* __restrict__ ws) {
  __shared__ float lds_cb[CHUNK * LDS_STRIDE];
  __shared__ float lds_nrm[CHUNK];

  const int tid  = threadIdx.x;
  const int wave = tid >> 5;
  const int lane = tid & 31;
  const int wt   = blockIdx.x * 4 + wave;      // 16-row tile id (exact: 375*4 = 1500)
  const int mrow = lane & 15;
  const int khalf = (lane >> 4) << 1;          // 0 for lanes 0-15, 2 for lanes 16-31

  // A-matrix rows live at flat n = wt*16 + mrow; z is [B, D, T]
  const int n  = wt * 16 + mrow;
  const int bb = n / TLEN;
  const int tt = n - bb * TLEN;
  const float* zrow = z + (size_t)bb * DIM * TLEN + tt;

  // Load full 16x128 A tile into registers in WMMA A layout (ISA 7.12.2):
  // lane = mrow + 16*h holds K = 4s + 2h (+0/+1 across the v2f)
  v2f a[32];
  float pn = 0.0f;
#pragma unroll
  for (int s = 0; s < 32; ++s) {
    float x0 = zrow[(size_t)(4 * s + khalf) * TLEN];
    float x1 = zrow[(size_t)(4 * s + khalf + 1) * TLEN];
    a[s].x = x0; a[s].y = x1;
    pn += x0 * x0 + x1 * x1;
  }
  const float rownorm = pn + __shfl_xor(pn, 16, 32);   // ||z_row||^2, row = lane&15

  float mind[8];
  int   amin[8];
#pragma unroll
  for (int r = 0; r < 8; ++r) { mind[r] = 3.4e38f; amin[r] = 0; }

  const int col4  = tid & 31;   // float4 column for cooperative load
  const int rbase = tid >> 5;   // 0..3

  for (int chunk = 0; chunk < K_CODES / CHUNK; ++chunk) {
    const int base = chunk * CHUNK;
    // stage 64x128 codebook chunk into LDS (coalesced float4, padded stride)
#pragma unroll
    for (int j = 0; j < CHUNK / 4; ++j) {
      int row = rbase + j * 4;
      float4 v = *(const float4*)(cb + (size_t)(base + row) * DIM + col4 * 4);
      *(float4*)(lds_cb + row * LDS_STRIDE + col4 * 4) = v;
    }
    if (tid < CHUNK) lds_nrm[tid] = ws[WS_NRM + base + tid];
    if (chunk + 1 < K_CODES / CHUNK)
      __builtin_prefetch(cb + (size_t)(base + CHUNK) * DIM + tid * 64, 0, 0);
    __syncthreads();

#pragma unroll
    for (int sub = 0; sub < CHUNK / 16; ++sub) {
      v8f c = {0.f, 0.f, 0.f, 0.f, 0.f, 0.f, 0.f, 0.f};
      // B-matrix (4x16): lane = ncol + 16*h holds K = 4s + 2h (+0/+1)
      const float* bcol = lds_cb + (sub * 16 + mrow) * LDS_STRIDE + khalf;
#pragma unroll
      for (int s = 0; s < 32; ++s) {
        v2f b;
        b.x = bcol[4 * s];
        b.y = bcol[4 * s + 1];
        c = __builtin_amdgcn_wmma_f32_16x16x4_f32(false, a[s], false, b,
                                                  (short)0, c, false, false);
      }
      const float cn   = lds_nrm[sub * 16 + mrow];     // ||c||^2 for this lane's code
      const int   code = base + sub * 16 + mrow;
#pragma unroll
      for (int r = 0; r < 8; ++r) {
        float dd = cn - 2.0f * c[r];                   // dist - ||z||^2 (const/row)
        if (dd < mind[r]) { mind[r] = dd; amin[r] = code; }
      }
    }
    __syncthreads();
  }

  // reduce (min, argmin) across each 16-lane half; first-index tie-break
#pragma unroll
  for (int r = 0; r < 8; ++r) {
#pragma unroll
    for (int off = 8; off; off >>= 1) {
      float om = __shfl_xor(mind[r], off, 32);
      int   oi = __shfl_xor(amin[r], off, 32);
      if (om < mind[r] || (om == mind[r] && oi < amin[r])) { mind[r] = om; amin[r] = oi; }
    }
  }
  // sum of ||z_row||^2 over the 16 rows of this tile (same in every lane of a half)
  float nt = rownorm;
#pragma unroll
  for (int off = 8; off; off >>= 1) nt += __shfl_xor(nt, off, 32);

  int* widx = (int*)ws;  // WS_IDX == 0
  if (lane == 0 || lane == 16) {
    const int rb = (lane >> 4) * 8;          // lane0 -> rows 0-7, lane16 -> rows 8-15
    float lsum = (lane == 0) ? nt : 0.0f;
#pragma unroll
    for (int r = 0; r < 8; ++r) {
      widx[wt * 16 + rb + r] = amin[r];
      lsum += mind[r];                        // ||z||^2 + ||c||^2 - 2 z.c summed
    }
    ws[WS_LOSSP + wt * 2 + (lane >> 4)] = lsum;
  }
}

// ---------------- scatter: z_q / codes outputs + EMA sums & counts ----------------
__global__ void scatter_kernel(const float* __restrict__ z,
                               const float* __restrict__ cb,
                               float* __restrict__ out,
                               float* __restrict__ ws) {
  const int* widx = (const int*)ws;
  int t = blockIdx.x * 256 + threadIdx.x;
  if (t >= TLEN) return;
  int bd = blockIdx.y;                 // 0..B*D-1
  int b = bd >> 7, d = bd & 127;
  int n = b * TLEN + t;
  int idx = widx[n];
  size_t zoff = (size_t)b * DIM * TLEN + (size_t)d * TLEN + t;
  float zv = z[zoff];
  float q  = cb[(size_t)idx * DIM + d];
  out[OUT_ZQ + zoff] = zv + (q - zv);                  // straight-through, matches ref fp
  atomicAdd(ws + WS_SUM + (size_t)idx * DIM + d, zv);  // segment_sum
  if (d == 0) {
    atomicAdd(ws + WS_CNT + idx, 1.0f);                // integer-valued -> exact
    out[OUT_CODES + n] = (float)idx;
  }
}

// ---------------- threefry2x32 (jax PRNG) ----------------
__device__ __forceinline__ unsigned rotl32(unsigned x, int r) {
  return (x << r) | (x >> (32 - r));
}
__device__ void threefry2x32(unsigned k0, unsigned k1, unsigned& x0, unsigned& x1) {
  const int r0[4] = {13, 15, 26, 6}, r1[4] = {17, 29, 16, 24};
  unsigned k2 = k0 ^ k1 ^ 0x1BD11BDAu;
  x0 += k0; x1 += k1;
#pragma unroll
  for (int i = 0; i < 4; ++i) { x0 += x1; x1 = rotl32(x1, r0[i]); x1 ^= x0; }
  x0 += k1; x1 += k2 + 1;
#pragma unroll
  for (int i = 0; i < 4; ++i) { x0 += x1; x1 = rotl32(x1, r1[i]); x1 ^= x0; }
  x0 += k2; x1 += k0 + 2;
#pragma unroll
  for (int i = 0; i < 4; ++i) { x0 += x1; x1 = rotl32(x1, r0[i]); x1 ^= x0; }
  x0 += k0; x1 += k1 + 3;
#pragma unroll
  for (int i = 0; i < 4; ++i) { x0 += x1; x1 = rotl32(x1, r1[i]); x1 ^= x0; }
  x0 += k1; x1 += k2 + 4;
#pragma unroll
  for (int i = 0; i < 4; ++i) { x0 += x1; x1 = rotl32(x1, r0[i]); x1 ^= x0; }
  x0 += k2; x1 += k0 + 5;
}

// ---------------- EMA pass 1: new_cs, n-sum, dead flag, rand indices ----------------
__global__ void ema1_kernel(const float* __restrict__ ema_cs,
                            float* __restrict__ ws, float* __restrict__ out) {
  int k = blockIdx.x * 256 + threadIdx.x;
  float ncs = ema_cs[k] * 0.99f + ws[WS_CNT + k] * 0.01f;
  ws[WS_CSPD + k] = ncs;
  atomicAdd(ws + WS_NSUM, ncs);                      // n = sum(new_cs)
  out[OUT_CS + k] = (ncs < 2.0f) ? 1.0f : ncs;       // dead -> 1.0
  if (k < K_CODES / 2) {                             // jax key(42) -> (0, 42)
    unsigned x0 = (unsigned)k, x1 = (unsigned)(k + K_CODES / 2);
    threefry2x32(0u, 42u, x0, x1);
    int* rnd = (int*)(ws + WS_RAND);
    rnd[k]               = (int)(x0 % (unsigned)NVEC);
    rnd[k + K_CODES / 2] = (int)(x1 % (unsigned)NVEC);
  }
}

// ---------------- EMA pass 2: embed / dead resample ----------------
__global__ void ema2_kernel(const float* __restrict__ z,
                            const float* __restrict__ ema_ea,
                            const float* __restrict__ ws, float* __restrict__ out) {
  int gid = blockIdx.x * 256 + threadIdx.x;
  int k = gid >> 7, d = gid & 127;
  float nea  = ema_ea[gid] * 0.99f + ws[WS_SUM + gid] * 0.01f;
  float ncs  = ws[WS_CSPD + k];
  float nsum = ws[WS_NSUM];
  float cs = (ncs + 1e-5f) / (nsum + K_CODES * 1e-5f) * nsum;
  cs = fmaxf(cs, 1e-12f);
  float embed = nea / cs;
  const int* rnd = (const int*)(ws + WS_RAND);
  int ridx = rnd[k];
  int b = ridx / TLEN, t = ridx - b * TLEN;
  float sampled = z[(size_t)b * DIM * TLEN + (size_t)d * TLEN + t];
  bool dead = ncs < 2.0f;
  out[OUT_CB + gid] = dead ? sampled : embed;
  out[OUT_EA + gid] = dead ? sampled : nea;
}

// ---------------- deterministic loss reduction ----------------
__global__ void loss_kernel(const float* __restrict__ ws, float* __restrict__ out) {
  __shared__ float red[256];
  float s = 0.0f;
  for (int i = threadIdx.x; i < 2 * NTILES; i += 256) s += ws[WS_LOSSP + i];
  red[threadIdx.x] = s;
  __syncthreads();
  for (int o = 128; o; o >>= 1) {
    if ((int)threadIdx.x < o) red[threadIdx.x] += red[threadIdx.x + o];
    __syncthreads();
  }
  if (threadIdx.x == 0) out[OUT_LOSS] = red[0] / (float)(NVEC * DIM);
}

extern "C" void kernel_launch(void* const* d_in, const int* in_sizes, int n_in,
                              void* d_out, int out_size, void* d_ws, size_t ws_size,
                              hipStream_t stream) {
  const float* z      = (const float*)d_in[0];
  const float* cb     = (const float*)d_in[1];
  const float* ema_cs = (const float*)d_in[2];
  const float* ema_ea = (const float*)d_in[3];
  float* out = (float*)d_out;
  float* ws  = (float*)d_ws;

  int zeroN = K_CODES + K_CODES * DIM;
  zero_ws_kernel<<<(zeroN + 255) / 256, 256, 0, stream>>>(ws);
  norms_kernel<<<K_CODES / 4, 128, 0, stream>>>(cb, ws);
  vq_kernel<<<NTILES / 4, 128, 0, stream>>>(z, cb, ws);
  scatter_kernel<<<dim3((TLEN + 255) / 256, BATCH * DIM), 256, 0, stream>>>(z, cb, out, ws);
  ema1_kernel<<<K_CODES / 256, 256, 0, stream>>>(ema_cs, ws, out);
  ema2_kernel<<<(K_CODES * DIM) / 256, 256, 0, stream>>>(z, ema_ea, ws, out);
  loss_kernel<<<1, 256, 0, stream>>>(ws, out);
}